// FlowModel_20461224198351
// MI455X (gfx1250) — compile-verified
//
#include <hip/hip_runtime.h>
#include <cstdint>
#include <cstddef>

typedef __attribute__((ext_vector_type(2))) float v2f;
typedef __attribute__((ext_vector_type(8))) float v8f;

namespace {
constexpr int NI = 8, H = 512, W = 512, CE = 32;
constexpr int KTOT = 3 * 3 * CE;          // 288
constexpr float GAMMA_FLOW = 0.001f;
constexpr float GAMMA_MASK = 0.0005f;
constexpr float ALPHA = 0.5f;
constexpr size_t NPIX = (size_t)NI * H * W;
}

// ---------------------------------------------------------------------------
// Kernel A: 3x3 conv (32 -> 3) + bias + tanh using V_WMMA_F32_16X16X4_F32.
// Block = 128 threads (4 wave32), covers 64 pixels along x of one row y.
// A (16x4 fp32): lanes 0-15 = M, v0/v1 = K,K+1 ; lanes 16-31 = K+2,K+3.
// B (4x16 fp32): lane n = output col; same K-half split as A.
// C/D (16x16 fp32): VGPR r = rows r (lanes 0-15) and r+8 (lanes 16-31).
// ---------------------------------------------------------------------------
__global__ __launch_bounds__(128) void conv_tanh_wmma(
    const float* __restrict__ encoded,   // (N,H,W,32)
    const float* __restrict__ conv_w,    // (3,3,32,3) HWIO flat: k*3 + o
    const float* __restrict__ conv_b,    // (3,)
    float* __restrict__ flow_out,        // (N,H,W,2)
    float* __restrict__ mask_out)        // (N,H,W)
{
    __shared__ float tile[3 * 66 * CE];   // rows y-1..y+1, cols x0-1..x0+64
    __shared__ float wT[16 * KTOT];       // transposed, zero-padded weights

    const int tid  = threadIdx.x;
    const int x0   = blockIdx.x * 64;
    const int y    = blockIdx.y;
    const int nimg = blockIdx.z;

    // Stage encoded halo tile (zero 'SAME' padding at borders).
    for (int i = tid; i < 3 * 66 * CE; i += 128) {
        int r   = i / (66 * CE);
        int rem = i - r * (66 * CE);
        int col = rem / CE;
        int c   = rem - col * CE;
        int gy  = y + r - 1;
        int gx  = x0 + col - 1;
        float v = 0.0f;
        if (gy >= 0 && gy < H && gx >= 0 && gx < W)
            v = encoded[(((size_t)nimg * H + gy) * W + gx) * CE + c];
        tile[i] = v;
    }
    // Stage weights, transposed to [n][k] and padded to 16 output columns.
    for (int i = tid; i < 16 * KTOT; i += 128) {
        int n = i / KTOT;
        int k = i - n * KTOT;
        wT[i] = (n < 3) ? conv_w[k * 3 + n] : 0.0f;
    }
    __syncthreads();

    const int lane = tid & 31;
    const int wave = tid >> 5;
    const int m    = lane & 15;   // A row (pixel in 16-tile) / B,D column
    const int half = lane >> 4;   // K-half selector (0: K,K+1 ; 1: K+2,K+3)
    const int lx   = wave * 16 + m;

    v8f acc = {};
    #pragma unroll
    for (int ky = 0; ky < 3; ++ky) {
        #pragma unroll
        for (int kx = 0; kx < 3; ++kx) {
            const float* arow = &tile[(ky * 66 + lx + kx) * CE];
            const float* brow = &wT[m * KTOT + (ky * 3 + kx) * CE];
            #pragma unroll
            for (int c0 = 0; c0 < CE; c0 += 4) {
                v2f a = *(const v2f*)(arow + c0 + 2 * half);
                v2f b = *(const v2f*)(brow + c0 + 2 * half);
                acc = __builtin_amdgcn_wmma_f32_16x16x4_f32(
                    false, a, false, b, (short)0, acc, false, false);
            }
        }
    }

    // D layout: lane holds column m for pixels (r + 8*half), r = 0..7.
    if (m < 3) {
        const float bn = conv_b[m];
        #pragma unroll
        for (int r = 0; r < 8; ++r) {
            int px = x0 + wave * 16 + (r + 8 * half);
            size_t pix = ((size_t)nimg * H + y) * W + px;
            float v = tanhf(acc[r] + bn);
            if (m < 2) flow_out[pix * 2 + m] = v;
            else       mask_out[pix] = v;
        }
    }
}

// ---------------------------------------------------------------------------
// Kernel B: dual bilinear warp + mask blend (exact reference math).
// ---------------------------------------------------------------------------
__device__ __forceinline__ void bilin3(const float* __restrict__ img, int choff,
                                       float coh, float cow, float out[3])
{
    float hh = (coh + 1.0f) * (float)(H - 1) * 0.5f;
    float ww = (cow + 1.0f) * (float)(W - 1) * 0.5f;
    int h0 = (int)floorf(hh);
    int w0 = (int)floorf(ww);
    int h1 = min(max(h0 + 1, 0), H - 1);
    int w1 = min(max(w0 + 1, 0), W - 1);
    h0 = min(max(h0, 0), H - 1);
    w0 = min(max(w0, 0), W - 1);
    float h1f = (float)h1, w1f = (float)w1;
    float dru = (h1f - hh) * (w1f - ww);
    float drd = (1.0f - (h1f - hh)) * (w1f - ww);
    float dlu = (h1f - hh) * (1.0f - (w1f - ww));
    float dld = (1.0f - (h1f - hh)) * (1.0f - (w1f - ww));
    const float* pru = img + ((size_t)h0 * W + w0) * 6 + choff;
    const float* prd = img + ((size_t)h1 * W + w0) * 6 + choff;
    const float* plu = img + ((size_t)h0 * W + w1) * 6 + choff;
    const float* pld = img + ((size_t)h1 * W + w1) * 6 + choff;
    #pragma unroll
    for (int c = 0; c < 3; ++c)
        out[c] = dru * pru[c] + drd * prd[c] + dlu * plu[c] + dld * pld[c];
}

__global__ __launch_bounds__(256) void warp_combine(
    const float* __restrict__ frames,    // (N,H,W,6)
    const float* __restrict__ flow,      // (N,H,W,2)
    const float* __restrict__ mask,      // (N,H,W)
    float* __restrict__ stacked)         // (N,H,W,3)
{
    size_t p = (size_t)blockIdx.x * blockDim.x + threadIdx.x;
    if (p >= NPIX) return;
    int n   = (int)(p / ((size_t)H * W));
    int rem = (int)(p - (size_t)n * H * W);
    int y = rem / W;
    int x = rem - y * W;

    float f0 = flow[p * 2 + 0];
    float f1 = flow[p * 2 + 1];
    float mk = mask[p];

    float hh = -1.0f + 2.0f * (float)y / (float)(H - 1);
    float ww = -1.0f + 2.0f * (float)x / (float)(W - 1);

    float ch0 = hh + f0 * ALPHA;
    float cw0 = ww + f1 * ALPHA;
    float ch1 = hh - f0 * (1.0f - ALPHA);
    float cw1 = ww - f1 * (1.0f - ALPHA);

    const float* img = frames + (size_t)n * H * W * 6;
    float o0[3], o1[3];
    bilin3(img, 0, ch0, cw0, o0);
    bilin3(img, 3, ch1, cw1, o1);

    float m = 0.5f * (1.0f + mk);
    #pragma unroll
    for (int c = 0; c < 3; ++c)
        stacked[p * 3 + c] = m * o0[c] + (1.0f - m) * o1[c];
}

// ---------------------------------------------------------------------------
// Kernels C/D: deterministic two-stage TV reduction.
// ---------------------------------------------------------------------------
__global__ __launch_bounds__(256) void tv_partial(
    const float* __restrict__ flow, const float* __restrict__ mask,
    float* __restrict__ partial)
{
    __shared__ float red[256];
    int tid = threadIdx.x;
    size_t p = (size_t)blockIdx.x * 256 + tid;
    float s = 0.0f;
    if (p < NPIX) {
        int rem = (int)(p % ((size_t)H * W));
        int y = rem / W;
        int x = rem - y * W;
        float a0 = flow[p * 2 + 0];
        float a1 = flow[p * 2 + 1];
        float mv = mask[p];
        float sf = 0.0f, sm = 0.0f;
        if (y < H - 1) {
            sf += fabsf(flow[(p + W) * 2 + 0] - a0)
                + fabsf(flow[(p + W) * 2 + 1] - a1);
            sm += fabsf(mask[p + W] - mv);
        }
        if (x < W - 1) {
            sf += fabsf(flow[(p + 1) * 2 + 0] - a0)
                + fabsf(flow[(p + 1) * 2 + 1] - a1);
            sm += fabsf(mask[p + 1] - mv);
        }
        s = GAMMA_FLOW * sf + GAMMA_MASK * sm;
    }
    red[tid] = s;
    __syncthreads();
    for (int o = 128; o > 0; o >>= 1) {
        if (tid < o) red[tid] += red[tid + o];
        __syncthreads();
    }
    if (tid == 0) partial[blockIdx.x] = red[0];
}

__global__ __launch_bounds__(256) void tv_final(
    const float* __restrict__ partial, int nblocks, float* __restrict__ out)
{
    __shared__ float red[256];
    int tid = threadIdx.x;
    float s = 0.0f;
    for (int i = tid; i < nblocks; i += 256) s += partial[i];
    red[tid] = s;
    __syncthreads();
    for (int o = 128; o > 0; o >>= 1) {
        if (tid < o) red[tid] += red[tid + o];
        __syncthreads();
    }
    if (tid == 0) out[0] = red[0] / (float)(NPIX * 3);
}

// ---------------------------------------------------------------------------
extern "C" void kernel_launch(void* const* d_in, const int* in_sizes, int n_in,
                              void* d_out, int out_size, void* d_ws, size_t ws_size,
                              hipStream_t stream)
{
    const float* frames  = (const float*)d_in[0];
    const float* encoded = (const float*)d_in[1];
    const float* conv_w  = (const float*)d_in[2];
    const float* conv_b  = (const float*)d_in[3];

    float* out     = (float*)d_out;
    float* flow    = out + NPIX * 3;            // flow region of d_out
    float* tv      = out + NPIX * 3 + NPIX * 2; // final scalar
    float* mask    = (float*)d_ws;              // NPIX floats
    float* partial = mask + NPIX;               // reduction partials

    dim3 gA(W / 64, H, NI);
    conv_tanh_wmma<<<gA, 128, 0, stream>>>(encoded, conv_w, conv_b, flow, mask);

    int nb = (int)((NPIX + 255) / 256);
    warp_combine<<<nb, 256, 0, stream>>>(frames, flow, mask, out);
    tv_partial<<<nb, 256, 0, stream>>>(flow, mask, partial);
    tv_final<<<1, 256, 0, stream>>>(partial, nb, tv);
}